// CTCLayer_60773787239110
// MI455X (gfx1250) — compile-verified
//
#include <hip/hip_runtime.h>

// Problem constants (from reference setup_inputs)
constexpr int B = 256;
constexpr int T = 512;
constexpr int L = 128;
constexpr int C = 128;
constexpr int S = 2 * L + 1;        // 257 extended states
constexpr int BLANK = C - 1;        // 127
constexpr float NEGINF = -1e30f;

// ---------------------------------------------------------------------------
// CDNA5 async global->LDS helpers (builtin if available, else inline asm)
// ---------------------------------------------------------------------------
#if defined(__has_builtin)
# if __has_builtin(__builtin_amdgcn_global_load_async_to_lds_b128)
#  define HAVE_ASYNC_B128_BUILTIN 1
# endif
# if __has_builtin(__builtin_amdgcn_s_wait_asynccnt)
#  define HAVE_WAIT_ASYNC_BUILTIN 1
# endif
#endif

typedef int v4i __attribute__((vector_size(16)));

__device__ __forceinline__ void async_copy_b128(const float* gsrc, float* ldst) {
#ifdef HAVE_ASYNC_B128_BUILTIN
  // Prototype (from hipcc diagnostic): (v4i addrspace(1)*, v4i addrspace(3)*, Ii, Ii)
  __builtin_amdgcn_global_load_async_to_lds_b128(
      (__attribute__((address_space(1))) v4i*)(void*)gsrc,
      (__attribute__((address_space(3))) v4i*)(void*)ldst,
      /*offset=*/0, /*cpol=*/0);
#else
  unsigned laddr =
      (unsigned)(unsigned long long)(__attribute__((address_space(3))) void*)(void*)ldst;
  asm volatile("global_load_async_to_lds_b128 %0, %1, off"
               :
               : "v"(laddr), "v"(gsrc)
               : "memory");
#endif
}

__device__ __forceinline__ void wait_async0() {
#ifdef HAVE_WAIT_ASYNC_BUILTIN
  __builtin_amdgcn_s_wait_asynccnt(0);
#else
  asm volatile("s_wait_asynccnt 0" ::: "memory");
#endif
}

// ---------------------------------------------------------------------------
// Kernel 1: logZ[b,t] = logsumexp over C of y_pred[b,t,:]   (wave per row)
// ---------------------------------------------------------------------------
__global__ void __launch_bounds__(256) logz_kernel(const float* __restrict__ yp,
                                                   float* __restrict__ logZ) {
  const int row  = (blockIdx.x * blockDim.x + threadIdx.x) >> 5;  // wave id
  const int lane = threadIdx.x & 31;
  if (row >= B * T) return;

  const float4 v = ((const float4*)(yp + (size_t)row * C))[lane];  // 16B/lane

  float m = fmaxf(fmaxf(v.x, v.y), fmaxf(v.z, v.w));
  #pragma unroll
  for (int off = 16; off > 0; off >>= 1) m = fmaxf(m, __shfl_xor(m, off, 32));

  float s = __expf(v.x - m) + __expf(v.y - m) + __expf(v.z - m) + __expf(v.w - m);
  #pragma unroll
  for (int off = 16; off > 0; off >>= 1) s += __shfl_xor(s, off, 32);

  if (lane == 0) logZ[row] = m + __logf(s);
}

// ---------------------------------------------------------------------------
// Kernel 2: CTC forward scan. One block per batch element, thread = state s.
// alpha ping-pong in LDS (1 barrier/step); logits row double-buffered in LDS,
// prefetched one step ahead by wave 0 via async global->LDS.
// ---------------------------------------------------------------------------
__global__ void __launch_bounds__(288, 1) ctc_scan_kernel(
    const float* __restrict__ yp, const int* __restrict__ yt,
    const float* __restrict__ logZ, float* __restrict__ loss) {
  __shared__ float alpha[2][S + 3];                 // +2 NEG pad at the front
  __shared__ __align__(16) float rowbuf[2][C];
  __shared__ int sh_len;

  const int b = blockIdx.x;
  const int s = threadIdx.x;                        // state index (valid s < S)
  const float* logits = yp + (size_t)b * T * C;
  const float* lz     = logZ + (size_t)b * T;
  const int*   lbl    = yt + (size_t)b * L;

  // ext[s] and skip_ok[s] in registers
  int  ext_s = BLANK;
  bool skip  = false;
  if (s < S && (s & 1)) {
    const int k   = s >> 1;
    const int cur = lbl[k];
    ext_s = cur;
    if (s >= 3) skip = (cur != lbl[k - 1]);
  }

  // label_length = #nonzero labels (deterministic integer atomic)
  if (threadIdx.x == 0) sh_len = 0;
  __syncthreads();
  if (s < L && lbl[s] != 0) atomicAdd(&sh_len, 1);

  // alpha_0 (uses row t=0 directly; only states 0 and 1 are live)
  if (s == 0) {
    alpha[0][0] = alpha[0][1] = NEGINF;
    alpha[1][0] = alpha[1][1] = NEGINF;
  }
  if (s < S) {
    float a0 = NEGINF;
    if (s == 0)      a0 = logits[BLANK] - lz[0];
    else if (s == 1) a0 = logits[ext_s] - lz[0];
    alpha[0][2 + s] = a0;
  }

  // Prefetch row t=1 into rowbuf[1] (wave 0: 32 lanes x 16B = 512B row)
  if (s < 32) async_copy_b128(logits + C + s * 4, &rowbuf[1][s * 4]);

  for (int t = 1; t < T; ++t) {
    const int cur = t & 1;
    if (s < 32) wait_async0();       // row t landed in rowbuf[cur]
    __syncthreads();                 // publish row t + alpha[t-1]; old reads done

    // Safe to overwrite rowbuf[cur^1] now: all of its readers passed the barrier
    if (s < 32 && t + 1 < T)
      async_copy_b128(logits + (size_t)(t + 1) * C + s * 4, &rowbuf[cur ^ 1][s * 4]);

    if (s < S) {
      const float a  = alpha[cur ^ 1][2 + s];
      const float a1 = alpha[cur ^ 1][1 + s];
      const float a2 = skip ? alpha[cur ^ 1][s] : NEGINF;
      const float m  = fmaxf(a, fmaxf(a1, a2));
      const float sum = __expf(a - m) + __expf(a1 - m) + __expf(a2 - m);
      const float comb = m + __logf(sum);
      alpha[cur][2 + s] = comb + rowbuf[cur][ext_s] - lz[t];
    }
  }

  __syncthreads();
  if (s == 0) {
    const int l = sh_len;                       // l >= 1 by construction
    const float f0 = alpha[(T - 1) & 1][2 + 2 * l];
    const float f1 = alpha[(T - 1) & 1][2 + 2 * l - 1];
    const float m  = fmaxf(f0, f1);
    loss[b] = -(m + __logf(__expf(f0 - m) + __expf(f1 - m)));
  }
}

// ---------------------------------------------------------------------------
// Kernel 3: deterministic mean over B losses -> d_out[0]
// ---------------------------------------------------------------------------
__global__ void __launch_bounds__(256) mean_kernel(const float* __restrict__ loss,
                                                   float* __restrict__ out) {
  __shared__ float sh[B];
  const int tid = threadIdx.x;
  sh[tid] = loss[tid];
  __syncthreads();
  #pragma unroll
  for (int off = B / 2; off > 0; off >>= 1) {
    if (tid < off) sh[tid] += sh[tid + off];
    __syncthreads();
  }
  if (tid == 0) out[0] = sh[0] * (1.0f / (float)B);
}

// ---------------------------------------------------------------------------
extern "C" void kernel_launch(void* const* d_in, const int* in_sizes, int n_in,
                              void* d_out, int out_size, void* d_ws, size_t ws_size,
                              hipStream_t stream) {
  const int*   y_true = (const int*)d_in[0];    // [B, L] int32
  const float* y_pred = (const float*)d_in[1];  // [B, T, C] float32
  float* out = (float*)d_out;                   // scalar mean loss

  float* logZ = (float*)d_ws;                   // B*T floats (512 KB)
  float* loss = logZ + (size_t)B * T;           // B floats

  (void)in_sizes; (void)n_in; (void)out_size; (void)ws_size;

  const int rows = B * T;                       // 131072
  logz_kernel<<<rows / 8, 256, 0, stream>>>(y_pred, logZ);
  ctc_scan_kernel<<<B, 288, 0, stream>>>(y_pred, y_true, logZ, loss);
  mean_kernel<<<1, B, 0, stream>>>(loss, out);
}